// CTCLayer_8229157339443
// MI455X (gfx1250) — compile-verified
//
#include <hip/hip_runtime.h>
#include <hip/hip_bf16.h>

#define Bn 256
#define Tn 1024
#define Cn 128
#define Ln 64
#define Sn 129            // 2*L+1
#define BLANKC 127
#define EPSf 1e-7f
#define NEGf (-1e30f)

// 4 x i32 vector, as required by the async-to-LDS b128 builtin
typedef int v4i __attribute__((vector_size(16)));
typedef v4i __attribute__((address_space(1))) *gv4p;  // global (AS1)
typedef v4i __attribute__((address_space(3))) *lv4p;  // LDS (AS3)

#if __has_builtin(__builtin_amdgcn_global_load_async_to_lds_b128)
#define HAVE_ASYNC 1
#else
#define HAVE_ASYNC 0
#endif

#if __has_builtin(__builtin_amdgcn_s_wait_asynccnt)
#define WAIT_ASYNC(n) __builtin_amdgcn_s_wait_asynccnt(n)
#else
#define WAIT_ASYNC(n) asm volatile("s_wait_asynccnt " #n)
#endif

// One workgroup per batch element. 160 threads = 5 wave32.
// Threads 0..128 own alpha states; wave 0 (threads 0..31) drives the async
// global->LDS pipeline of probability rows (512B/row, 3 rows in flight).
__global__ __launch_bounds__(160) void ctc_loss_kernel(
    const int* __restrict__ y_true,    // [B, L]
    const float* __restrict__ y_pred,  // [B, T, C] softmax probs
    float* __restrict__ out)           // [B, 1]
{
    __shared__ float rowbuf[4][Cn];     // 4-deep ring of probability rows
    __shared__ float abuf[2][Sn + 3];   // double-buffered alpha exchange
    __shared__ int   labels[Ln];

    const int tid = threadIdx.x;
    const int b   = blockIdx.x;
    const float* yp = y_pred + (size_t)b * Tn * Cn;

    if (tid < Ln) labels[tid] = y_true[b * Ln + tid];

#if HAVE_ASYNC
    // Prologue: start rows 0 and 1 streaming into LDS (ASYNCcnt = 2).
    if (tid < 32) {
        __builtin_amdgcn_global_load_async_to_lds_b128(
            (gv4p)(yp + (size_t)0 * Cn + tid * 4),
            (lv4p)(&rowbuf[0][tid * 4]), 0, 0);
        __builtin_amdgcn_global_load_async_to_lds_b128(
            (gv4p)(yp + (size_t)1 * Cn + tid * 4),
            (lv4p)(&rowbuf[1][tid * 4]), 0, 0);
    }
#endif

    __syncthreads();  // labels visible

    // Per-state constants (extended sequence: blank,l0,blank,l1,...,blank).
    int ext_c   = BLANKC;
    int canskip = 0;
    if (tid < Sn) {
        if (tid & 1) {
            ext_c = labels[tid >> 1];                       // label index (s-1)/2
            if (tid >= 3) canskip = (ext_c != labels[(tid - 2) >> 1]);
        }
    }

    float a = NEGf;  // this thread's alpha[s]

    for (int t = 0; t < Tn; ++t) {
        // Publish alpha from step t-1 for neighbor reads (double buffered, so
        // one barrier per step is race-free).
        if (t > 0 && tid < Sn) abuf[t & 1][tid] = a;

#if HAVE_ASYNC
        if (tid < 32) {
            if (t + 2 < Tn) {
                __builtin_amdgcn_global_load_async_to_lds_b128(
                    (gv4p)(yp + (size_t)(t + 2) * Cn + tid * 4),
                    (lv4p)(&rowbuf[(t + 2) & 3][tid * 4]), 0, 0);
            }
            // Ensure row t has landed: outstanding after issue is
            // rows t..min(t+2,T-1); allow all but the oldest.
            if (t < Tn - 2)        WAIT_ASYNC(2);
            else if (t == Tn - 2)  WAIT_ASYNC(1);
            else                   WAIT_ASYNC(0);
        }
#else
        if (tid < Cn) rowbuf[t & 3][tid] = yp[(size_t)t * Cn + tid];
#endif

        __syncthreads();  // row t + alpha(t-1) visible to all waves

        if (tid < Sn) {
            const float lp = __logf(rowbuf[t & 3][ext_c] + EPSf);
            if (t == 0) {
                a = (tid <= 1) ? lp : NEGf;  // only s=0,1 reachable at t=0
            } else {
                const float s1 = (tid >= 1) ? abuf[t & 1][tid - 1] : NEGf;
                const float s2 = canskip    ? abuf[t & 1][tid - 2] : NEGf;
                float m = fmaxf(a, fmaxf(s1, s2));
                float sum = __expf(a - m) + __expf(s1 - m) + __expf(s2 - m);
                a = m + __logf(sum) + lp;
            }
        }
    }

    // loss = -logaddexp(alpha[S-1], alpha[S-2])
    if (tid < Sn) abuf[0][tid] = a;
    __syncthreads();
    if (tid == 0) {
        const float x = abuf[0][Sn - 1];
        const float y = abuf[0][Sn - 2];
        const float m = fmaxf(x, y);
        out[b] = -(m + __logf(__expf(x - m) + __expf(y - m)));
    }
}

extern "C" void kernel_launch(void* const* d_in, const int* in_sizes, int n_in,
                              void* d_out, int out_size, void* d_ws, size_t ws_size,
                              hipStream_t stream) {
    const int*   y_true = (const int*)d_in[0];    // [256, 64] int32
    const float* y_pred = (const float*)d_in[1];  // [256, 1024, 128] f32
    float*       out    = (float*)d_out;          // [256, 1] f32

    ctc_loss_kernel<<<dim3(Bn), dim3(160), 0, stream>>>(y_true, y_pred, out);
}